// GNN_regresion_54932631716440
// MI455X (gfx1250) — compile-verified
//
#include <hip/hip_runtime.h>
#include <math.h>

typedef float v2f __attribute__((ext_vector_type(2)));
typedef float v8f __attribute__((ext_vector_type(8)));

__device__ __forceinline__ float gelu_f(float x) {
    // jax.nn.gelu default (approximate=True, tanh form)
    float x3 = x * x * x;
    float t  = tanhf(0.7978845608028654f * (x + 0.044715f * x3));
    return 0.5f * x * (1.0f + t);
}

__global__ void k_zero(float* __restrict__ p, int n) {
    int i = blockIdx.x * blockDim.x + threadIdx.x;
    if (i < n) p[i] = 0.0f;
}

// deg[col] += ew  (self-loop +1 added later)
__global__ void k_deg(const int* __restrict__ col, const float* __restrict__ ew,
                      float* __restrict__ deg, int E) {
    int e = blockIdx.x * blockDim.x + threadIdx.x;
    if (e < E) atomicAdd(&deg[col[e]], ew[e]);
}

// in-place: deg -> dinv = rsqrt(deg + 1)
__global__ void k_dinv(float* __restrict__ deg, int N) {
    int i = blockIdx.x * blockDim.x + threadIdx.x;
    if (i < N) {
        float d = deg[i] + 1.0f;
        deg[i] = (d > 0.0f) ? rsqrtf(d) : 0.0f;
    }
}

// norm[e] = dinv[row]*ew*dinv[col];  layer-1 scalar agg: s1[col] += norm*x[row]
__global__ void k_norm_s1(const int* __restrict__ row, const int* __restrict__ col,
                          const float* __restrict__ ew, const float* __restrict__ x,
                          const float* __restrict__ dinv, float* __restrict__ norm,
                          float* __restrict__ s1, int E) {
    int e = blockIdx.x * blockDim.x + threadIdx.x;
    if (e < E) {
        int r = row[e], c = col[e];
        float nm = dinv[r] * ew[e] * dinv[c];
        norm[e] = nm;
        atomicAdd(&s1[c], nm * x[r]);
    }
}

// h1[i,f] = gelu((s1[i] + dinv[i]^2 * x[i]) * W1[f] + b1[f])
__global__ void k_h1(const float* __restrict__ s1, const float* __restrict__ dinv,
                     const float* __restrict__ x, const float* __restrict__ W1,
                     const float* __restrict__ b1, float* __restrict__ h1, int N) {
    int t = blockIdx.x * blockDim.x + threadIdx.x;
    if (t < N * 64) {
        int i = t >> 6, f = t & 63;
        float di = dinv[i];
        float s = s1[i] + di * di * x[i];
        h1[t] = gelu_f(s * W1[f] + b1[f]);
    }
}

// D[N,64] = A[N,64] @ W[64,64] via V_WMMA_F32_16X16X4_F32.
// One block = 128 threads = 4 waves; block does 16 rows, each wave one 16x16 tile.
__global__ __launch_bounds__(128)
void k_gemm64_wmma(const float* __restrict__ A, const float* __restrict__ W,
                   float* __restrict__ D, int N) {
    __shared__ float sW[64 * 64];   // 16 KB
    __shared__ float sA[16 * 65];   // padded stride 65 -> conflict-free fragment reads
    int tid = threadIdx.x;
    int r0  = blockIdx.x * 16;

    for (int idx = tid; idx < 64 * 64; idx += 128) sW[idx] = W[idx];
    for (int idx = tid; idx < 16 * 64; idx += 128) {
        int r = idx >> 6, k = idx & 63;
        int gr = r0 + r;
        sA[r * 65 + k] = (gr < N) ? A[(size_t)gr * 64 + k] : 0.0f;
    }
    __syncthreads();

    int wave = tid >> 5;
    int lane = tid & 31;
    int half = lane >> 4;    // 0: K+0/K+1 ; 1: K+2/K+3
    int l16  = lane & 15;
    int c0   = wave * 16;

    v8f acc = {0.f, 0.f, 0.f, 0.f, 0.f, 0.f, 0.f, 0.f};
#pragma unroll
    for (int k0 = 0; k0 < 64; k0 += 4) {
        int kk = k0 + half * 2;
        v2f a, b;
        // A 16x4: lane l16 = row M; VGPR0 = K(kk), VGPR1 = K(kk+1)
        a.x = sA[l16 * 65 + kk];
        a.y = sA[l16 * 65 + kk + 1];
        // B 4x16: row K striped across lanes; VGPR0 = K(kk), VGPR1 = K(kk+1)
        b.x = sW[kk * 64 + c0 + l16];
        b.y = sW[(kk + 1) * 64 + c0 + l16];
        acc = __builtin_amdgcn_wmma_f32_16x16x4_f32(
            false, a, false, b, (short)0, acc, false, false);
    }

    // C/D: VGPR j -> M = j + half*8, N = c0 + l16
#pragma unroll
    for (int j = 0; j < 8; ++j) {
        int gr = r0 + half * 8 + j;
        if (gr < N) D[(size_t)gr * 64 + c0 + l16] = acc[j];
    }
}

// layer-2 edge scatter: agg2[col,f] += xw2[row,f] * norm[e]; 1 thread per (e,f)
__global__ void k_agg2(const int* __restrict__ row, const int* __restrict__ col,
                       const float* __restrict__ norm, const float* __restrict__ xw2,
                       float* __restrict__ agg2, long long total) {
    long long t = (long long)blockIdx.x * blockDim.x + threadIdx.x;
    if (t < total) {
        int e = (int)(t >> 6);
        int f = (int)(t & 63);
        atomicAdd(&agg2[(size_t)col[e] * 64 + f], xw2[(size_t)row[e] * 64 + f] * norm[e]);
    }
}

// h2 = gelu(agg2 + dinv^2*xw2 + b2); fused mean-pool accumulation
__global__ void k_h2_pool(const float* __restrict__ agg2, const float* __restrict__ xw2,
                          const float* __restrict__ dinv, const float* __restrict__ b2,
                          const int* __restrict__ batch, float* __restrict__ gsum,
                          float* __restrict__ gcnt, int N) {
    int t = blockIdx.x * blockDim.x + threadIdx.x;
    if (t < N * 64) {
        int i = t >> 6, f = t & 63;
        float di = dinv[i];
        float v = agg2[t] + di * di * xw2[t] + b2[f];
        float h = gelu_f(v);
        int g = batch[i];
        atomicAdd(&gsum[g * 64 + f], h);
        if (f == 0) atomicAdd(&gcnt[g], 1.0f);
    }
}

// head: out[g] = fc2_b + gelu(mean_g @ fc1_W + fc1_b) @ fc2_W
__global__ void k_head(const float* __restrict__ gsum, const float* __restrict__ gcnt,
                       const float* __restrict__ fc1W, const float* __restrict__ fc1b,
                       const float* __restrict__ fc2W, const float* __restrict__ fc2b,
                       float* __restrict__ out, int G) {
    int g = blockIdx.x * blockDim.x + threadIdx.x;
    if (g >= G) return;
    float inv = 1.0f / fmaxf(gcnt[g], 1.0f);
    float v[64];
#pragma unroll
    for (int k = 0; k < 64; ++k) v[k] = gsum[g * 64 + k] * inv;
    float o = fc2b[0];
    for (int j = 0; j < 32; ++j) {
        float a = fc1b[j];
#pragma unroll
        for (int k = 0; k < 64; ++k) a += v[k] * fc1W[k * 32 + j];
        o += gelu_f(a) * fc2W[j];
    }
    out[g] = o;
}

extern "C" void kernel_launch(void* const* d_in, const int* in_sizes, int n_in,
                              void* d_out, int out_size, void* d_ws, size_t ws_size,
                              hipStream_t stream) {
    const float* x    = (const float*)d_in[0];
    const int*   ei   = (const int*)d_in[1];   // [2, E] (int32: jax default x64 off)
    const float* ea   = (const float*)d_in[2];
    const int*   bat  = (const int*)d_in[3];
    const float* W1   = (const float*)d_in[4];
    const float* b1   = (const float*)d_in[5];
    const float* W2   = (const float*)d_in[6];
    const float* b2   = (const float*)d_in[7];
    const float* fc1W = (const float*)d_in[8];
    const float* fc1b = (const float*)d_in[9];
    const float* fc2W = (const float*)d_in[10];
    const float* fc2b = (const float*)d_in[11];

    const int N = in_sizes[0];       // 100000
    const int E = in_sizes[2];       // 3200000
    const int G = out_size;          // 256
    const int* row = ei;
    const int* col = ei + E;

    float* ws = (float*)d_ws;
    size_t o = 0;
    float* dinv = ws + o; o += (size_t)N;        // degree, then rsqrt in-place
    float* s1   = ws + o; o += (size_t)N;        // layer-1 scalar aggregate
    float* agg2 = ws + o; o += (size_t)N * 64;   // layer-2 aggregate
    float* gsum = ws + o; o += (size_t)G * 64;   // pooled sums
    float* gcnt = ws + o; o += (size_t)G;        // pooled counts
    size_t zcount = o;                           // zero-init everything above
    float* norm = ws + o; o += (size_t)E;        // per-edge sym-norm weight
    float* h1   = ws + o; o += (size_t)N * 64;
    float* xw2  = ws + o; o += (size_t)N * 64;

    const int B = 256;
    k_zero<<<(int)((zcount + B - 1) / B), B, 0, stream>>>(ws, (int)zcount);
    k_deg<<<(E + B - 1) / B, B, 0, stream>>>(col, ea, dinv, E);
    k_dinv<<<(N + B - 1) / B, B, 0, stream>>>(dinv, N);
    k_norm_s1<<<(E + B - 1) / B, B, 0, stream>>>(row, col, ea, x, dinv, norm, s1, E);

    int nf = N * 64;
    k_h1<<<(nf + B - 1) / B, B, 0, stream>>>(s1, dinv, x, W1, b1, h1, N);

    k_gemm64_wmma<<<(N + 15) / 16, 128, 0, stream>>>(h1, W2, xw2, N);

    long long tot = (long long)E * 64;
    k_agg2<<<(int)((tot + B - 1) / B), B, 0, stream>>>(row, col, norm, xw2, agg2, tot);

    k_h2_pool<<<(nf + B - 1) / B, B, 0, stream>>>(agg2, xw2, dinv, b2, bat, gsum, gcnt, N);
    k_head<<<(G + B - 1) / B, B, 0, stream>>>(gsum, gcnt, fc1W, fc1b, fc2W, fc2b,
                                              (float*)d_out, G);
}